// PointNextPartDecoder_31061203485302
// MI455X (gfx1250) — compile-verified
//
#include <hip/hip_runtime.h>
#include <hip/hip_bf16.h>

typedef float v2f __attribute__((ext_vector_type(2)));
typedef float v8f __attribute__((ext_vector_type(8)));

#define BN_EPS 1e-5f
#define BATCH 16

// ---------------------------------------------------------------------------
// Class-embedding: cls_feat[b][o] = relu(convc_w[o][label[b]] + convc_b[o])
// ---------------------------------------------------------------------------
__global__ void cls_embed_kernel(const int* __restrict__ label,
                                 const float* __restrict__ w,   // [64,16]
                                 const float* __restrict__ bias,// [64]
                                 float* __restrict__ out)       // [B,64]
{
    int b = blockIdx.x;
    int o = threadIdx.x; // 64 threads
    int l = label[b];
    out[b * 64 + o] = fmaxf(w[o * 16 + l] + bias[o], 0.0f);
}

// ---------------------------------------------------------------------------
// Copy f_hi [B,C,N] into concat buffer [B,Ctot,N] at channel offset coff
// ---------------------------------------------------------------------------
__global__ void copy_into_kernel(const float* __restrict__ src,
                                 float* __restrict__ dst,
                                 int C, int N, int Ctot, int coff, int total)
{
    int i = blockIdx.x * blockDim.x + threadIdx.x;
    if (i >= total) return;
    int n = i % N;
    int c = (i / N) % C;
    int b = i / (N * C);
    dst[((size_t)b * Ctot + coff + c) * N + n] = src[i];
}

// ---------------------------------------------------------------------------
// Broadcast cls_feat [B,64] along N into concat buffer channels [0,64)
// ---------------------------------------------------------------------------
__global__ void broadcast_cls_kernel(const float* __restrict__ cls,
                                     float* __restrict__ dst,
                                     int N, int Ctot, int total)
{
    int i = blockIdx.x * blockDim.x + threadIdx.x;
    if (i >= total) return;
    int n = i % N;
    int c = (i / N) % 64;
    int b = i / (N * 64);
    dst[((size_t)b * Ctot + c) * N + n] = cls[b * 64 + c];
}

// ---------------------------------------------------------------------------
// 3-NN inverse-distance interpolation.
//   Phi: [B,N,3] query points, Plo: [B,M,3], Flo: [B,Clo,M]
//   Writes into concat buffer [B,Ctot,N] at channel offset coff.
// One thread per (b,n). Plo staged in LDS (M*3 floats).
// ---------------------------------------------------------------------------
__global__ void knn3_interp_kernel(const float* __restrict__ Phi,
                                   const float* __restrict__ Plo,
                                   const float* __restrict__ Flo,
                                   float* __restrict__ Out,
                                   int N, int M, int Clo, int Ctot, int coff)
{
    extern __shared__ float sP[]; // M*3 floats
    const int b = blockIdx.y;
    const int n = blockIdx.x * blockDim.x + threadIdx.x;

    const float* plo = Plo + (size_t)b * M * 3;
    for (int i = threadIdx.x; i < M * 3; i += blockDim.x) sP[i] = plo[i];
    __syncthreads();

    const float* q = Phi + ((size_t)b * N + n) * 3;
    const float qx = q[0], qy = q[1], qz = q[2];

    float d0 = 3.4e38f, d1 = 3.4e38f, d2 = 3.4e38f;
    int   i0 = 0,       i1 = 0,       i2 = 0;
    for (int m = 0; m < M; ++m) {
        float dx = qx - sP[m * 3 + 0];
        float dy = qy - sP[m * 3 + 1];
        float dz = qz - sP[m * 3 + 2];
        float d  = dx * dx + dy * dy + dz * dz;
        if (d < d0)      { d2 = d1; i2 = i1; d1 = d0; i1 = i0; d0 = d; i0 = m; }
        else if (d < d1) { d2 = d1; i2 = i1; d1 = d;  i1 = m; }
        else if (d < d2) { d2 = d;  i2 = m; }
    }
    float w0 = 1.0f / (d0 + 1e-8f);
    float w1 = 1.0f / (d1 + 1e-8f);
    float w2 = 1.0f / (d2 + 1e-8f);
    float ws = 1.0f / (w0 + w1 + w2);
    w0 *= ws; w1 *= ws; w2 *= ws;

    const float* fb = Flo + (size_t)b * Clo * M;
    float* ob = Out + ((size_t)b * Ctot + coff) * N + n;
    for (int c = 0; c < Clo; ++c) {
        const float* fr = fb + (size_t)c * M;
        ob[(size_t)c * N] = w0 * fr[i0] + w1 * fr[i1] + w2 * fr[i2];
    }
}

// ---------------------------------------------------------------------------
// 1x1 conv + inference BN + ReLU as fp32 WMMA GEMM.
//   X: [B,Cin,N], W: [Cout,Cin], Y: [B,Cout,N]
// One wave32 per 16(M=Cout) x 64(N) output tile; K loop in steps of 4 using
// V_WMMA_F32_16X16X4_F32.
//   A 16x4 layout: lanes 0-15 = M rows; VGPR0/1 = {K0,K1} (lanes<16) or
//                  {K2,K3} (lanes>=16).
//   B 4x16 layout (mirror): lanes 0-15 = N cols; lane half selects K pair.
//   C/D 16x16 f32: lane<16 -> N=lane, M=r; lane>=16 -> N=lane-16, M=8+r.
// ---------------------------------------------------------------------------
__global__ __launch_bounds__(32)
void gemm_bn_relu_wmma(const float* __restrict__ X,
                       const float* __restrict__ W,
                       const float* __restrict__ Bs,  // conv bias
                       const float* __restrict__ Gm,  // gamma
                       const float* __restrict__ Bt,  // beta
                       const float* __restrict__ Mn,  // running mean
                       const float* __restrict__ Vr,  // running var
                       float* __restrict__ Y,
                       int Cin, int Cout, int N)
{
    const int lane = threadIdx.x;
    const int ln   = lane & 15;
    const int hi   = lane >> 4;           // 0 or 1 -> K pair select
    const int n0   = blockIdx.x * 64;
    const int m0   = blockIdx.y * 16;
    const int b    = blockIdx.z;

    const float* Wrow = W + (size_t)(m0 + ln) * Cin + 2 * hi;
    const float* Xcol = X + (size_t)b * Cin * N + (size_t)(2 * hi) * N + n0 + ln;

    v8f z = {};
    v8f acc0 = z, acc1 = z, acc2 = z, acc3 = z;

    for (int k = 0; k < Cin; k += 4) {
        v2f a = *(const v2f*)(Wrow + k);      // W[m][k+2hi], W[m][k+2hi+1]
        const float* xk = Xcol + (size_t)k * N;
        v2f b0; b0.x = xk[0];      b0.y = xk[N];
        v2f b1; b1.x = xk[16];     b1.y = xk[N + 16];
        v2f b2; b2.x = xk[32];     b2.y = xk[N + 32];
        v2f b3; b3.x = xk[48];     b3.y = xk[N + 48];
        acc0 = __builtin_amdgcn_wmma_f32_16x16x4_f32(false, a, false, b0,
                                                     (short)0, acc0, false, false);
        acc1 = __builtin_amdgcn_wmma_f32_16x16x4_f32(false, a, false, b1,
                                                     (short)0, acc1, false, false);
        acc2 = __builtin_amdgcn_wmma_f32_16x16x4_f32(false, a, false, b2,
                                                     (short)0, acc2, false, false);
        acc3 = __builtin_amdgcn_wmma_f32_16x16x4_f32(false, a, false, b3,
                                                     (short)0, acc3, false, false);
    }

    float* Yb = Y + (size_t)b * Cout * N;
#pragma unroll
    for (int r = 0; r < 8; ++r) {
        int c = m0 + r + 8 * hi;
        float s  = Gm[c] * rsqrtf(Vr[c] + BN_EPS);
        float sh = Bt[c] - Mn[c] * s + Bs[c] * s;   // folded bias+BN shift
        float* yp = Yb + (size_t)c * N + n0 + ln;
        yp[0]  = fmaxf(acc0[r] * s + sh, 0.0f);
        yp[16] = fmaxf(acc1[r] * s + sh, 0.0f);
        yp[32] = fmaxf(acc2[r] * s + sh, 0.0f);
        yp[48] = fmaxf(acc3[r] * s + sh, 0.0f);
    }
}

// ---------------------------------------------------------------------------
extern "C" void kernel_launch(void* const* d_in, const int* in_sizes, int n_in,
                              void* d_out, int out_size, void* d_ws, size_t ws_size,
                              hipStream_t stream)
{
    const float* p1 = (const float*)d_in[0];
    const float* p2 = (const float*)d_in[1];
    const float* p3 = (const float*)d_in[2];
    const float* p4 = (const float*)d_in[3];
    const float* p5 = (const float*)d_in[4];
    const float* f1 = (const float*)d_in[5];
    const float* f2 = (const float*)d_in[6];
    const float* f3 = (const float*)d_in[7];
    const float* f4 = (const float*)d_in[8];
    const float* f5 = (const float*)d_in[9];
    const int*   lab    = (const int*)d_in[10];
    const float* convcw = (const float*)d_in[11];
    const float* convcb = (const float*)d_in[12];

    // fp_params: stage s, layer l, tensor t (w,b,gamma,beta,mean,var)
    auto P = [&](int s, int l, int t) -> const float* {
        return (const float*)d_in[13 + s * 12 + l * 6 + t];
    };

    // Workspace layout (floats)
    float* ws   = (float*)d_ws;
    float* cls  = ws;                    // 16*64          = 1024
    float* f4n  = cls + 1024;            // 16*512*64      = 524288
    float* f3n  = f4n + 524288;          // 16*256*256     = 1048576
    float* f2n  = f3n + 1048576;         // 16*128*1024    = 2097152
    float* cat  = f2n + 2097152;         // 16*256*4096    = 16777216 (max, reused)
    float* mid  = cat + 16777216;        // 16*64*4096     = 4194304  (max, reused)

    // class embedding
    cls_embed_kernel<<<BATCH, 64, 0, stream>>>(lab, convcw, convcb, cls);

    // ---------------- Stage A: f4n = FP(p4,f4, p5,f5)  [1536 -> 512 -> 512, N=64]
    {
        int C = 512, N = 64, M = 16, Clo = 1024, Ctot = 1536;
        int tot = BATCH * C * N;
        copy_into_kernel<<<(tot + 255) / 256, 256, 0, stream>>>(f4, cat, C, N, Ctot, 0, tot);
        knn3_interp_kernel<<<dim3(N / 64, BATCH), 64, M * 3 * sizeof(float), stream>>>(
            p4, p5, f5, cat, N, M, Clo, Ctot, C);
        gemm_bn_relu_wmma<<<dim3(N / 64, 512 / 16, BATCH), 32, 0, stream>>>(
            cat, P(0,0,0), P(0,0,1), P(0,0,2), P(0,0,3), P(0,0,4), P(0,0,5),
            mid, 1536, 512, N);
        gemm_bn_relu_wmma<<<dim3(N / 64, 512 / 16, BATCH), 32, 0, stream>>>(
            mid, P(0,1,0), P(0,1,1), P(0,1,2), P(0,1,3), P(0,1,4), P(0,1,5),
            f4n, 512, 512, N);
    }

    // ---------------- Stage B: f3n = FP(p3,f3, p4,f4n) [768 -> 256 -> 256, N=256]
    {
        int C = 256, N = 256, M = 64, Clo = 512, Ctot = 768;
        int tot = BATCH * C * N;
        copy_into_kernel<<<(tot + 255) / 256, 256, 0, stream>>>(f3, cat, C, N, Ctot, 0, tot);
        knn3_interp_kernel<<<dim3(N / 64, BATCH), 64, M * 3 * sizeof(float), stream>>>(
            p3, p4, f4n, cat, N, M, Clo, Ctot, C);
        gemm_bn_relu_wmma<<<dim3(N / 64, 256 / 16, BATCH), 32, 0, stream>>>(
            cat, P(1,0,0), P(1,0,1), P(1,0,2), P(1,0,3), P(1,0,4), P(1,0,5),
            mid, 768, 256, N);
        gemm_bn_relu_wmma<<<dim3(N / 64, 256 / 16, BATCH), 32, 0, stream>>>(
            mid, P(1,1,0), P(1,1,1), P(1,1,2), P(1,1,3), P(1,1,4), P(1,1,5),
            f3n, 256, 256, N);
    }

    // ---------------- Stage C: f2n = FP(p2,f2, p3,f3n) [384 -> 128 -> 128, N=1024]
    {
        int C = 128, N = 1024, M = 256, Clo = 256, Ctot = 384;
        int tot = BATCH * C * N;
        copy_into_kernel<<<(tot + 255) / 256, 256, 0, stream>>>(f2, cat, C, N, Ctot, 0, tot);
        knn3_interp_kernel<<<dim3(N / 64, BATCH), 64, M * 3 * sizeof(float), stream>>>(
            p2, p3, f3n, cat, N, M, Clo, Ctot, C);
        gemm_bn_relu_wmma<<<dim3(N / 64, 128 / 16, BATCH), 32, 0, stream>>>(
            cat, P(2,0,0), P(2,0,1), P(2,0,2), P(2,0,3), P(2,0,4), P(2,0,5),
            mid, 384, 128, N);
        gemm_bn_relu_wmma<<<dim3(N / 64, 128 / 16, BATCH), 32, 0, stream>>>(
            mid, P(2,1,0), P(2,1,1), P(2,1,2), P(2,1,3), P(2,1,4), P(2,1,5),
            f2n, 128, 128, N);
    }

    // ---------------- Stage D: out = FP(p1,[cls;f1], p2,f2n) [256 -> 64 -> 64, N=4096]
    {
        int N = 4096, M = 1024, Clo = 128, Ctot = 256;
        int totc = BATCH * 64 * N;
        broadcast_cls_kernel<<<(totc + 255) / 256, 256, 0, stream>>>(cls, cat, N, Ctot, totc);
        int totf = BATCH * 64 * N;
        copy_into_kernel<<<(totf + 255) / 256, 256, 0, stream>>>(f1, cat, 64, N, Ctot, 64, totf);
        knn3_interp_kernel<<<dim3(N / 64, BATCH), 64, M * 3 * sizeof(float), stream>>>(
            p1, p2, f2n, cat, N, M, Clo, Ctot, 128);
        gemm_bn_relu_wmma<<<dim3(N / 64, 64 / 16, BATCH), 32, 0, stream>>>(
            cat, P(3,0,0), P(3,0,1), P(3,0,2), P(3,0,3), P(3,0,4), P(3,0,5),
            mid, 256, 64, N);
        gemm_bn_relu_wmma<<<dim3(N / 64, 64 / 16, BATCH), 32, 0, stream>>>(
            mid, P(3,1,0), P(3,1,1), P(3,1,2), P(3,1,3), P(3,1,4), P(3,1,5),
            (float*)d_out, 64, 64, N);
    }
}